// Capsule_21663815041426
// MI455X (gfx1250) — compile-verified
//
#include <hip/hip_runtime.h>

// Capsule graph routing (NGM-style), 2 layers, for MI455X (gfx1250, wave32).
//
// Strategy:
//  - gather table (N*128 f32 = 25.6MB) is L2-resident (192MB L2) -> random
//    neighbor gathers are cheap; do them ONCE per layer into registers.
//  - one node per wave32: lane = k*4 + s owns float4 of capsule k.
//    z[16 neighbors] lives in 64 VGPRs/lane across all 6 routing iterations.
//  - ALL cross-lane reductions use VALU DPP / v_permlanex16 (no DS pipe,
//    no s_wait_dscnt on the critical dependency chain):
//      xor1 : quad_perm(1,0,3,2) = 0xB1
//      xor2 : quad_perm(2,3,0,1) = 0x4E
//      xor4 : ROW_XMASK(4)      = 0x164
//      xor8 : ROW_XMASK(8)      = 0x168
//      xor16: v_permlanex16 (identity selects -> swap 16-lane rows)
//  - softmax over k: |logit| <= 1 (unit-norm z rows and u, Cauchy-Schwarz),
//    so the max-subtraction pass is mathematically redundant -> skip it.
//  - FC 128x128 done with V_WMMA_F32_16X16X4_F32 (full-precision f32 WMMA),
//    with relu + per-capsule L2-normalization FUSED into its epilogue: a
//    16-wide column tile == one capsule, and each D-matrix row lives in a
//    16-lane row of the wave -> capsule norm is a within-row DPP reduction.
//    This removes an entire 51MB normalize pass.

#define NHID   16
#define DFEAT  128
#define MNBR   16
#define ROUTIT 6

typedef __attribute__((ext_vector_type(2))) float v2f;
typedef __attribute__((ext_vector_type(8))) float v8f;

__device__ __forceinline__ float dot4(float4 a, float4 b) {
  return a.x * b.x + a.y * b.y + a.z * b.z + a.w * b.w;
}

// x + (x from lane^mask), executed entirely on the VALU (DPP).
template <int CTRL>
__device__ __forceinline__ float dpp_xor_add(float x) {
  int xi = __builtin_bit_cast(int, x);
  int yi = __builtin_amdgcn_update_dpp(0, xi, CTRL, 0xF, 0xF, true);
  return x + __builtin_bit_cast(float, yi);
}

// value from lane^16 via v_permlanex16_b32 (identity lane selects).
__device__ __forceinline__ float xor16_get(float x) {
  int xi = __builtin_bit_cast(int, x);
  int yi = __builtin_amdgcn_permlanex16(xi, xi, 0x76543210, 0xFEDCBA98,
                                        false, false);
  return __builtin_bit_cast(float, yi);
}

__device__ __forceinline__ float quad_pair_sum(float x) {   // over xor1, xor2
  return dpp_xor_add<0x4E>(dpp_xor_add<0xB1>(x));
}

__device__ __forceinline__ float kgroup_sum(float x) {      // over xor4,8,16
  float t = dpp_xor_add<0x168>(dpp_xor_add<0x164>(x));
  return t + xor16_get(t);
}

__device__ __forceinline__ float row16_sum(float x) {       // over xor1,2,4,8
  return dpp_xor_add<0x168>(
      dpp_xor_add<0x164>(dpp_xor_add<0x4E>(dpp_xor_add<0xB1>(x))));
}

__device__ __forceinline__ float caps_inv_norm(float n2) {
  return (n2 > 1e-24f) ? __builtin_amdgcn_rsqf(n2) : 1e12f;
}

// ---------------------------------------------------------------------------
// Per-capsule L2 normalization: one thread per capsule (16 contiguous floats).
// x / max(||x||, 1e-12)  ==  x * rsq(n2) for n2 > 1e-24, else x * 1e12.
// ---------------------------------------------------------------------------
__global__ __launch_bounds__(256) void normalize_caps_kernel(
    const float* __restrict__ in, float* __restrict__ out, long long ncaps) {
  long long gid = (long long)blockIdx.x * blockDim.x + threadIdx.x;
  if (gid >= ncaps) return;
  const float4* src = (const float4*)(in + gid * NHID);
  float4 v0 = src[0], v1 = src[1], v2 = src[2], v3 = src[3];
  float n2 = dot4(v0, v0) + dot4(v1, v1) + dot4(v2, v2) + dot4(v3, v3);
  float inv = caps_inv_norm(n2);
  v0.x *= inv; v0.y *= inv; v0.z *= inv; v0.w *= inv;
  v1.x *= inv; v1.y *= inv; v1.z *= inv; v1.w *= inv;
  v2.x *= inv; v2.y *= inv; v2.z *= inv; v2.w *= inv;
  v3.x *= inv; v3.y *= inv; v3.z *= inv; v3.w *= inv;
  float4* dst = (float4*)(out + gid * NHID);
  dst[0] = v0; dst[1] = v1; dst[2] = v2; dst[3] = v3;
}

// ---------------------------------------------------------------------------
// Dynamic routing, one node per wave32.
//   lane = k*4 + s   (k in [0,8) capsule, s in [0,4) quarter of the 16 dims)
// Each lane holds float4 slices; z (16 neighbors) stays in registers for all
// 6 iterations. Writes relu(u).
// ---------------------------------------------------------------------------
__global__ __launch_bounds__(256) void routing_relu_kernel(
    const float* __restrict__ xn, const int* __restrict__ nb,
    float* __restrict__ out, int n) {
  const int wave = threadIdx.x >> 5;
  const int lane = threadIdx.x & 31;
  const int node = blockIdx.x * 8 + wave;
  if (node >= n) return;                 // wave-uniform branch

  const int off = (lane >> 2) * NHID + (lane & 3) * 4;  // k*16 + s*4

  const float4 xn4 = *(const float4*)(xn + (size_t)node * DFEAT + off);

  // Gather 16 neighbors' slices into registers (indices are wave-uniform
  // -> compiler scalarizes nb loads; data gathers hit L2).
  float4 z[MNBR];
#pragma unroll
  for (int m = 0; m < MNBR; ++m) {
    int idx = nb[node * MNBR + m];
    if ((unsigned)idx < (unsigned)n) {
      z[m] = *(const float4*)(xn + (size_t)idx * DFEAT + off);
    } else {                              // reference's zero-pad row
      z[m] = float4{0.f, 0.f, 0.f, 0.f};
    }
  }

  float4 u = xn4;
#pragma unroll
  for (int it = 0; it < ROUTIT; ++it) {
    // logits[m][k] = sum_d z[m][k][d] * u[k][d]; reduce over the 4 s-lanes
    // (xor1, xor2 -> quad_perm DPP).
    float p[MNBR];
#pragma unroll
    for (int m = 0; m < MNBR; ++m) {
      p[m] = quad_pair_sum(dot4(z[m], u));  // replicated across s
    }
    // softmax over k (lanes differing in bits 2..4 -> ROW_XMASK + permlanex16).
    // |logit| <= 1 -> no max-subtraction needed.
#pragma unroll
    for (int m = 0; m < MNBR; ++m) {
      float e = __expf(p[m]);
      float ssum = kgroup_sum(e);
      p[m] = e * __builtin_amdgcn_rcpf(ssum);
    }
    // u[k][d] = sum_m z*p + xn
    float4 acc = xn4;
#pragma unroll
    for (int m = 0; m < MNBR; ++m) {
      acc.x += z[m].x * p[m];
      acc.y += z[m].y * p[m];
      acc.z += z[m].z * p[m];
      acc.w += z[m].w * p[m];
    }
    if (it < ROUTIT - 1) {
      float n2 = quad_pair_sum(dot4(acc, acc));
      float inv = caps_inv_norm(n2);
      u.x = acc.x * inv; u.y = acc.y * inv;
      u.z = acc.z * inv; u.w = acc.w * inv;
    } else {
      u = acc;
    }
  }

  float4 r;
  r.x = fmaxf(u.x, 0.f); r.y = fmaxf(u.y, 0.f);
  r.z = fmaxf(u.z, 0.f); r.w = fmaxf(u.w, 0.f);
  *(float4*)(out + (size_t)node * DFEAT + off) = r;
}

// ---------------------------------------------------------------------------
// Yn = caps_normalize(relu(H @ W^T + b)), H:[n,128], W:[128,128] row-major,
// via f32 WMMA. One 16x16 output tile per wave; 32 x V_WMMA_F32_16X16X4_F32.
// A (16x4, MxK): lanes 0-15 hold rows, {K0,K1}; lanes 16-31 hold {K2,K3}.
// B (4x16, KxN): B[k][o] = W[o][k], so a lane's B column is a row of W.
// C/D (16x16): VGPR r, lanes 0-15 -> M=r, lanes 16-31 -> M=r+8; N=lane&15.
// Epilogue: column tile colT == capsule colT (16 dims), and row M of D lives
// in one 16-lane row of the wave -> per-capsule squared norm = within-row
// DPP reduction (xor1,2,4,8). Fuses the layer-1 normalize pass.
// ---------------------------------------------------------------------------
__global__ __launch_bounds__(256) void fc_relu_capsnorm_wmma_kernel(
    const float* __restrict__ H, const float* __restrict__ W,
    const float* __restrict__ bias, float* __restrict__ Y, int n) {
  const int lane   = threadIdx.x & 31;
  const int wave   = threadIdx.x >> 5;
  const int waveId = blockIdx.x * 8 + wave;
  const int colT   = waveId & 7;          // 8 column tiles of 16 over D=128
  const int rowT   = waveId >> 3;         // row tile (block shares one rowT)
  const int l16    = lane & 15;
  const int hi     = lane >> 4;           // 0: K{0,1}  1: K{2,3}

  int row = rowT * 16 + l16;
  if (row >= n) row = n - 1;              // clamp loads; stores are guarded
  const int col = colT * 16 + l16;

  const float* hptr = H + (size_t)row * DFEAT + 2 * hi;
  const float* wptr = W + (size_t)col * DFEAT + 2 * hi;

  v8f acc = {0.f, 0.f, 0.f, 0.f, 0.f, 0.f, 0.f, 0.f};
#pragma unroll
  for (int kk = 0; kk < 32; ++kk) {       // K = 32 * 4 = 128
    v2f a = *(const v2f*)(hptr + kk * 4);
    v2f b = *(const v2f*)(wptr + kk * 4);
    acc = __builtin_amdgcn_wmma_f32_16x16x4_f32(
        /*neg_a=*/false, a, /*neg_b=*/false, b,
        /*c_mod=*/(short)0, acc, /*reuse_a=*/false, /*reuse_b=*/false);
  }

  const float bv = bias[col];
#pragma unroll
  for (int r = 0; r < 8; ++r) {
    float v = fmaxf(acc[r] + bv, 0.f);     // relu(h @ W^T + b)
    float n2 = row16_sum(v * v);           // capsule norm^2 over the 16 cols
    v *= caps_inv_norm(n2);                // x / max(||x||, 1e-12)
    int orow = rowT * 16 + r + 8 * hi;
    if (orow < n) {
      Y[(size_t)orow * DFEAT + col] = v;
    }
  }
}

// ---------------------------------------------------------------------------
// Pipeline:
//   1. normalize(x)                    -> wsA   (xn, layer 0)
//   2. routing(wsA) + relu             -> wsB   (h0)
//   3. wmma fc(wsB) + relu + capsnorm  -> wsA   (yn, layer 1)
//   4. routing(wsA) + relu             -> d_out (h1)
// Workspace: 2 * N * 128 floats = 51.2 MB.
// ---------------------------------------------------------------------------
extern "C" void kernel_launch(void* const* d_in, const int* in_sizes, int n_in,
                              void* d_out, int out_size, void* d_ws,
                              size_t ws_size, hipStream_t stream) {
  const float* x     = (const float*)d_in[0];
  const int*   nb    = (const int*)d_in[1];
  const float* fc1_w = (const float*)d_in[2];
  const float* fc1_b = (const float*)d_in[3];
  float* out = (float*)d_out;

  const int n = in_sizes[0] / DFEAT;      // 50000
  float* wsA = (float*)d_ws;
  float* wsB = wsA + (size_t)n * DFEAT;

  const long long ncaps = (long long)n * 8;
  const int normBlocks = (int)((ncaps + 255) / 256);
  const int routBlocks = (n + 7) / 8;     // 8 waves (nodes) per 256-thr block
  const int fcBlocks   = (n + 15) / 16;   // one row tile per block (8 waves)

  // Layer 0
  normalize_caps_kernel<<<normBlocks, 256, 0, stream>>>(x, wsA, ncaps);
  routing_relu_kernel<<<routBlocks, 256, 0, stream>>>(wsA, nb, wsB, n);
  // Layer 1 (fc + relu + per-capsule normalize fused)
  fc_relu_capsnorm_wmma_kernel<<<fcBlocks, 256, 0, stream>>>(wsB, fc1_w,
                                                             fc1_b, wsA, n);
  routing_relu_kernel<<<routBlocks, 256, 0, stream>>>(wsA, nb, out, n);
}